// QCNetMapEncoder_19799799234866
// MI455X (gfx1250) — compile-verified
//
#include <hip/hip_runtime.h>
#include <hip/hip_bf16.h>
#include <math.h>

#define H 128
#define Fq 64
#define NHD 8
#define TT 10
#define NPT 100000
#define NPL 10000
#define E1N 200000
#define E2N 160000

typedef float v2f __attribute__((ext_vector_type(2)));
typedef float v8f __attribute__((ext_vector_type(8)));

// ---------------------------------------------------------------------------
// CDNA5 fp32 WMMA: D(16x16) = A(16x4) x B(4x16) + C, full fp32 precision.
// A layout (32-bit, 16x4): lanes 0-15 -> rows M=lane, v0=K0,v1=K1;
//                          lanes 16-31 -> rows M=lane-16, v0=K2,v1=K3.
// B layout (symmetric): lanes 0-15: v0=row K0, v1=row K1 at col N=lane;
//                       lanes 16-31: v0=K2, v1=K3 at col N=lane-16.
// C/D layout: VGPR i, lane l -> M = i + 8*(l>>4), N = l&15.
// ---------------------------------------------------------------------------
__device__ inline v8f wmma_f32_k4(v2f a, v2f b, v8f c) {
  return __builtin_amdgcn_wmma_f32_16x16x4_f32(false, a, false, b, (short)0, c,
                                               false, false);
}

// Async global->LDS copy of one dword per lane (gfx1250 ASYNCcnt path).
__device__ inline void async_copy_b32(const float* gaddr, float* lds_ptr) {
  unsigned ldsoff = (unsigned)(size_t)lds_ptr;  // low 32 bits = LDS offset
  asm volatile("global_load_async_to_lds_b32 %0, %1, off" ::"v"(ldsoff),
               "v"(gaddr)
               : "memory");
}
__device__ inline void wait_async_all() {
  asm volatile("s_wait_asynccnt 0x0" ::: "memory");
}

// ============================ Fused Fourier MLP ============================
// out[n,:] = relu(LN(sum_d (relu(LN(feat(n,d) @ w1[d] + b1[d])) @ w2[d] + b2[d]))) @ wo + bo
template <int D>
__global__ void __launch_bounds__(64) fourier_kernel(
    const float* __restrict__ X, const float* __restrict__ freqs,
    const float* __restrict__ w1, const float* __restrict__ b1,
    const float* __restrict__ ln1g, const float* __restrict__ ln1b,
    const float* __restrict__ w2, const float* __restrict__ b2,
    const float* __restrict__ lnog, const float* __restrict__ lnob,
    const float* __restrict__ wo, const float* __restrict__ bo,
    float* __restrict__ out, int N) {
  __shared__ float sA[2][16 * 132];   // features (129 cols, padded to 132)
  __shared__ float sH[2][16 * H];     // hidden
  __shared__ float sAcc[2][16 * H];   // sum over d
  __shared__ float sX[2][16];
  const int wv = threadIdx.x >> 5;
  const int lane = threadIdx.x & 31;
  const int m0 = (blockIdx.x * 2 + wv) * 16;
  const int r15 = lane & 15;
  const int ko = (lane >> 4) * 2;

  // pull weights toward GL2 early (global_prefetch_b8)
  __builtin_prefetch(w1 + (size_t)lane * 512, 0, 1);
  __builtin_prefetch(w2 + (size_t)lane * 512, 0, 1);

  for (int idx = lane; idx < 16 * H; idx += 32) sAcc[wv][idx] = 0.f;

  for (int d = 0; d < D; ++d) {
    if (lane < 16) {
      int row = m0 + lane;
      if (row >= N) row = N - 1;
      sX[wv][lane] = X[(size_t)row * D + d];
    }
    __syncthreads();
    for (int idx = lane; idx < 16 * 132; idx += 32) {
      int r = idx / 132, j = idx - r * 132;
      float x = sX[wv][r];
      float val;
      if (j < Fq)
        val = cosf(6.2831853071795864f * x * freqs[d * Fq + j]);
      else if (j < 2 * Fq)
        val = sinf(6.2831853071795864f * x * freqs[d * Fq + j - Fq]);
      else if (j == 2 * Fq)
        val = x;
      else
        val = 0.f;
      sA[wv][idx] = val;
    }
    __syncthreads();
    // GEMM1: h1 = feat(16x129) @ w1[d](129x128) + b1[d]
    for (int nt = 0; nt < 8; ++nt) {
      int col = nt * 16 + r15;
      float bvv = b1[d * H + col];
      v8f acc;
      for (int i = 0; i < 8; ++i) acc[i] = bvv;
      for (int k0 = 0; k0 < 132; k0 += 4) {
        int ka = k0 + ko;
        v2f a = {sA[wv][r15 * 132 + ka], sA[wv][r15 * 132 + ka + 1]};
        v2f b;
        b.x = (ka < 129) ? w1[((size_t)d * 129 + ka) * H + col] : 0.f;
        b.y = (ka + 1 < 129) ? w1[((size_t)d * 129 + ka + 1) * H + col] : 0.f;
        acc = wmma_f32_k4(a, b, acc);
      }
      for (int i = 0; i < 8; ++i) {
        int M = i + ((lane >> 4) << 3);
        sH[wv][M * H + col] = acc[i];
      }
    }
    __syncthreads();
    // LN + relu rows of sH
    if (lane < 16) {
      float m = 0.f;
      for (int j = 0; j < H; ++j) m += sH[wv][lane * H + j];
      m *= (1.f / H);
      float v = 0.f;
      for (int j = 0; j < H; ++j) {
        float dd = sH[wv][lane * H + j] - m;
        v += dd * dd;
      }
      float inv = rsqrtf(v * (1.f / H) + 1e-5f);
      for (int j = 0; j < H; ++j) {
        float y = (sH[wv][lane * H + j] - m) * inv * ln1g[d * H + j] + ln1b[d * H + j];
        sH[wv][lane * H + j] = fmaxf(y, 0.f);
      }
    }
    __syncthreads();
    // GEMM2: sAcc += h @ w2[d] + b2[d]
    for (int nt = 0; nt < 8; ++nt) {
      int col = nt * 16 + r15;
      float bvv = b2[d * H + col];
      v8f acc;
      for (int i = 0; i < 8; ++i) acc[i] = bvv;
      for (int k0 = 0; k0 < H; k0 += 4) {
        int ka = k0 + ko;
        v2f a = {sH[wv][r15 * H + ka], sH[wv][r15 * H + ka + 1]};
        v2f b = {w2[((size_t)d * H + ka) * H + col],
                 w2[((size_t)d * H + ka + 1) * H + col]};
        acc = wmma_f32_k4(a, b, acc);
      }
      for (int i = 0; i < 8; ++i) {
        int M = i + ((lane >> 4) << 3);
        sAcc[wv][M * H + col] += acc[i];
      }
    }
    __syncthreads();
  }
  // final LN + relu into sH
  if (lane < 16) {
    float m = 0.f;
    for (int j = 0; j < H; ++j) m += sAcc[wv][lane * H + j];
    m *= (1.f / H);
    float v = 0.f;
    for (int j = 0; j < H; ++j) {
      float dd = sAcc[wv][lane * H + j] - m;
      v += dd * dd;
    }
    float inv = rsqrtf(v * (1.f / H) + 1e-5f);
    for (int j = 0; j < H; ++j) {
      float y = (sAcc[wv][lane * H + j] - m) * inv * lnog[j] + lnob[j];
      sH[wv][lane * H + j] = fmaxf(y, 0.f);
    }
  }
  __syncthreads();
  // GEMM3: out = h @ wo + bo
  for (int nt = 0; nt < 8; ++nt) {
    int col = nt * 16 + r15;
    float bvv = bo[col];
    v8f acc;
    for (int i = 0; i < 8; ++i) acc[i] = bvv;
    for (int k0 = 0; k0 < H; k0 += 4) {
      int ka = k0 + ko;
      v2f a = {sH[wv][r15 * H + ka], sH[wv][r15 * H + ka + 1]};
      v2f b = {wo[(size_t)ka * H + col], wo[(size_t)(ka + 1) * H + col]};
      acc = wmma_f32_k4(a, b, acc);
    }
    for (int i = 0; i < 8; ++i) {
      int M = i + ((lane >> 4) << 3);
      int row = m0 + M;
      if (row < N) out[(size_t)row * H + col] = acc[i];
    }
  }
}

// ===================== Generic WMMA GEMM: C = A@W (+bias, act) =============
// act: 0=none, 1=relu, 2=sigmoid
// A tile staged global->LDS via CDNA5 async-to-LDS (ASYNCcnt) path.
template <int K, int WPB>
__global__ void __launch_bounds__(32 * WPB) gemm_kernel(
    const float* __restrict__ A, const float* __restrict__ W,
    const float* __restrict__ bias, float* __restrict__ C, int M, int Ncols,
    int act) {
  __shared__ float sA[WPB][16 * K];
  const int wv = threadIdx.x >> 5;
  const int lane = threadIdx.x & 31;
  const int m0 = (blockIdx.x * WPB + wv) * 16;
  const int r15 = lane & 15;
  const int ko = (lane >> 4) * 2;

  // pull the (heavily reused, L2-resident) weight matrix toward GL2
  __builtin_prefetch(W + (size_t)threadIdx.x * 512, 0, 1);

  // async stage: 16 x K A-tile, one dword per lane per issue
  for (int idx = lane; idx < 16 * K; idx += 32) {
    int r = idx / K, k = idx - r * K;
    int row = m0 + r;
    if (row >= M) row = M - 1;
    async_copy_b32(&A[(size_t)row * K + k], &sA[wv][idx]);
  }
  wait_async_all();
  __syncthreads();

  for (int n0 = 0; n0 < Ncols; n0 += 16) {
    int col = n0 + r15;
    float bvv = bias ? bias[col] : 0.f;
    v8f acc;
    for (int i = 0; i < 8; ++i) acc[i] = bvv;
    for (int k0 = 0; k0 < K; k0 += 4) {
      int ka = k0 + ko;
      v2f a = {sA[wv][r15 * K + ka], sA[wv][r15 * K + ka + 1]};
      v2f b = {W[(size_t)ka * Ncols + col], W[(size_t)(ka + 1) * Ncols + col]};
      acc = wmma_f32_k4(a, b, acc);
    }
    for (int i = 0; i < 8; ++i) {
      float vl = acc[i];
      if (act == 1) vl = fmaxf(vl, 0.f);
      else if (act == 2) vl = 1.f / (1.f + expf(-vl));
      int row = m0 + i + ((lane >> 4) << 3);
      if (row < M) C[(size_t)row * Ncols + col] = vl;
    }
  }
}

// =============================== LayerNorm =================================
__global__ void ln_kernel(const float* __restrict__ x, const float* __restrict__ g,
                          const float* __restrict__ b, float* __restrict__ out,
                          int M) {
  int wv = (blockIdx.x * blockDim.x + threadIdx.x) >> 5;
  int lane = threadIdx.x & 31;
  if (wv >= M) return;
  const float* xr = x + (size_t)wv * H;
  float v0[4], s = 0.f;
  for (int i = 0; i < 4; ++i) {
    v0[i] = xr[lane + i * 32];
    s += v0[i];
  }
  for (int off = 16; off; off >>= 1) s += __shfl_xor(s, off, 32);
  float mean = s * (1.f / H);
  float vs = 0.f;
  for (int i = 0; i < 4; ++i) {
    float d = v0[i] - mean;
    vs += d * d;
  }
  for (int off = 16; off; off >>= 1) vs += __shfl_xor(vs, off, 32);
  float inv = rsqrtf(vs * (1.f / H) + 1e-5f);
  for (int i = 0; i < 4; ++i) {
    int j = lane + i * 32;
    out[(size_t)wv * H + j] = (v0[i] - mean) * inv * g[j] + b[j];
  }
}

// ============================ edge geometry ================================
__device__ inline float wrapang(float a) {
  const float PIv = 3.14159265358979323846f;
  float m = fmodf(a + PIv, 2.f * PIv);
  if (m < 0.f) m += 2.f * PIv;
  return -PIv + m;
}

__global__ void geom_kernel(const float* __restrict__ pos_s,
                            const float* __restrict__ or_s,
                            const float* __restrict__ pos_d,
                            const float* __restrict__ or_d,
                            const float* __restrict__ ovec_d,
                            const int* __restrict__ src,
                            const int* __restrict__ dst, float* __restrict__ r,
                            int E) {
  int e = blockIdx.x * blockDim.x + threadIdx.x;
  if (e >= E) return;
  int s = src[e], d = dst[e];
  float rx = pos_s[s * 2] - pos_d[d * 2];
  float ry = pos_s[s * 2 + 1] - pos_d[d * 2 + 1];
  float ro = wrapang(or_s[s] - or_d[d]);
  float cx = ovec_d[d * 2], cy = ovec_d[d * 2 + 1];
  r[(size_t)e * 3 + 0] = sqrtf(rx * rx + ry * ry);
  r[(size_t)e * 3 + 1] = atan2f(cx * ry - cy * rx, cx * rx + cy * ry);
  r[(size_t)e * 3 + 2] = ro;
}

// ===================== segment softmax (3 atomic passes) ===================
__device__ inline unsigned f2ord(float f) {
  unsigned u = __float_as_uint(f);
  return (u & 0x80000000u) ? ~u : (u | 0x80000000u);
}
__device__ inline float ord2f(unsigned u) {
  unsigned v = (u & 0x80000000u) ? (u & 0x7FFFFFFFu) : ~u;
  return __uint_as_float(v);
}

__global__ void sim_kernel(const float* __restrict__ q, const float* __restrict__ k,
                           const float* __restrict__ kr, const int* __restrict__ src,
                           const int* __restrict__ dst, unsigned* __restrict__ smax,
                           float* __restrict__ sim, int E) {
  int t = blockIdx.x * blockDim.x + threadIdx.x;
  if (t >= E * NHD) return;
  int e = t >> 3, h = t & 7;
  int s = src[e], d = dst[e];
  const float* qr = q + (size_t)d * H + h * 16;
  const float* kk = k + (size_t)s * H + h * 16;
  const float* kre = kr + (size_t)e * H + h * 16;
  float acc = 0.f;
  for (int i = 0; i < 16; ++i) acc += qr[i] * (kk[i] + kre[i]);
  acc *= 0.25f;  // HD^-0.5
  sim[t] = acc;
  atomicMax(&smax[(size_t)d * NHD + h], f2ord(acc));
}

__global__ void ex_kernel(float* __restrict__ sim, const unsigned* __restrict__ smax,
                          float* __restrict__ den, const int* __restrict__ dst,
                          int E) {
  int t = blockIdx.x * blockDim.x + threadIdx.x;
  if (t >= E * NHD) return;
  int e = t >> 3, h = t & 7;
  int d = dst[e];
  float exv = expf(sim[t] - ord2f(smax[(size_t)d * NHD + h]));
  sim[t] = exv;
  atomicAdd(&den[(size_t)d * NHD + h], exv);
}

__global__ void agg_kernel(const float* __restrict__ v, const float* __restrict__ vr,
                           const float* __restrict__ ex, const float* __restrict__ den,
                           const int* __restrict__ src, const int* __restrict__ dst,
                           float* __restrict__ agg, int E) {
  long long t = (long long)blockIdx.x * blockDim.x + threadIdx.x;
  if (t >= (long long)E * H) return;
  int e = (int)(t >> 7);
  int j = (int)(t & (H - 1));
  int h = j >> 4;
  int s = src[e], d = dst[e];
  float w = ex[(size_t)e * NHD + h] / (den[(size_t)d * NHD + h] + 1e-16f);
  atomicAdd(&agg[(size_t)d * H + j],
            (v[(size_t)s * H + j] + vr[(size_t)e * H + j]) * w);
}

// ============================ small elementwise ============================
__global__ void fill_f(float* p, float v, long long n) {
  long long t = (long long)blockIdx.x * blockDim.x + threadIdx.x;
  if (t < n) p[t] = v;
}
__global__ void fill_u(unsigned* p, unsigned v, long long n) {
  long long t = (long long)blockIdx.x * blockDim.x + threadIdx.x;
  if (t < n) p[t] = v;
}
__global__ void add_inplace(float* a, const float* b, long long n) {
  long long t = (long long)blockIdx.x * blockDim.x + threadIdx.x;
  if (t < n) a[t] += b[t];
}
__global__ void gate_kernel(float* agg, const float* g, const float* s, long long n) {
  long long t = (long long)blockIdx.x * blockDim.x + threadIdx.x;
  if (t < n) agg[t] = agg[t] + g[t] * (s[t] - agg[t]);
}
__global__ void cat_kernel(const float* a, const float* b, float* c, int M) {
  long long t = (long long)blockIdx.x * blockDim.x + threadIdx.x;
  if (t >= (long long)M * 2 * H) return;
  int row = (int)(t >> 8);
  int j = (int)(t & (2 * H - 1));
  c[t] = (j < H) ? a[(size_t)row * H + j] : b[(size_t)row * H + j - H];
}
__global__ void bcast_kernel(const float* x, float* out, long long n) {
  long long t = (long long)blockIdx.x * blockDim.x + threadIdx.x;
  if (t >= n) return;
  int j = (int)(t & (H - 1));
  long long row = (t >> 7) / TT;
  out[t] = x[row * H + j];
}

// ================================ host side ================================
static inline void gemm128(const float* A, const float* W, const float* b, float* C,
                           int M, int Ncols, int act, hipStream_t s) {
  gemm_kernel<128, 4><<<dim3((M + 63) / 64), dim3(128), 0, s>>>(A, W, b, C, M, Ncols, act);
}
static inline void gemm256(const float* A, const float* W, const float* b, float* C,
                           int M, int Ncols, int act, hipStream_t s) {
  gemm_kernel<256, 2><<<dim3((M + 31) / 32), dim3(64), 0, s>>>(A, W, b, C, M, Ncols, act);
}
static inline void gemm512(const float* A, const float* W, const float* b, float* C,
                           int M, int Ncols, int act, hipStream_t s) {
  gemm_kernel<512, 1><<<dim3((M + 15) / 16), dim3(32), 0, s>>>(A, W, b, C, M, Ncols, act);
}
static inline void lnrun(const float* x, const float* g, const float* b, float* out,
                         int M, hipStream_t s) {
  ln_kernel<<<dim3((M + 7) / 8), dim3(256), 0, s>>>(x, g, b, out, M);
}
#define PFp(i) ((const float*)d_in[(i)])
#define PIp(i) ((const int*)d_in[(i)])

static void run_fourier(void* const* d_in, int D, const float* X, int base,
                        float* out, int N, hipStream_t s) {
  dim3 grid((N + 31) / 32), block(64);
  if (D == 1)
    fourier_kernel<1><<<grid, block, 0, s>>>(X, PFp(base), PFp(base + 1), PFp(base + 2),
        PFp(base + 3), PFp(base + 4), PFp(base + 5), PFp(base + 6), PFp(base + 7),
        PFp(base + 8), PFp(base + 9), PFp(base + 10), out, N);
  else if (D == 2)
    fourier_kernel<2><<<grid, block, 0, s>>>(X, PFp(base), PFp(base + 1), PFp(base + 2),
        PFp(base + 3), PFp(base + 4), PFp(base + 5), PFp(base + 6), PFp(base + 7),
        PFp(base + 8), PFp(base + 9), PFp(base + 10), out, N);
  else
    fourier_kernel<3><<<grid, block, 0, s>>>(X, PFp(base), PFp(base + 1), PFp(base + 2),
        PFp(base + 3), PFp(base + 4), PFp(base + 5), PFp(base + 6), PFp(base + 7),
        PFp(base + 8), PFp(base + 9), PFp(base + 10), out, N);
}

// attn param order: 0 pn_src_g,1 pn_src_b,2 pn_r_g,3 pn_r_b,4 wq,5 bq,6 wk,7 wv,
// 8 bv,9 wkr,10 wvr,11 bvr,12 ws,13 bs,14 wg,15 bg,16 wo,17 bo,18 post_g,
// 19 post_b,20 ffn_g,21 ffn_b,22 w1,23 b1,24 w2,25 b2,[26 pn_dst_g,27 pn_dst_b]
static void run_attn_layer(void* const* d_in, int base, bool bip,
                           const float* xsrc_in, int nsrc, float* xpl, float* rn,
                           const int* src, const int* dst, int E, float* xsrcn,
                           float* xdstn, float* qbuf, float* kvbuf, float* ebuf,
                           float* aggb, float* t1, float* t2, float* catb,
                           float* ffnh, float* simb, unsigned* smax, float* denb,
                           hipStream_t s) {
  const float* gdst = bip ? PFp(base + 26) : PFp(base + 0);
  const float* bdst = bip ? PFp(base + 27) : PFp(base + 1);
  lnrun(xpl, gdst, bdst, xdstn, NPL, s);
  const float* src_act;
  if (bip) {
    lnrun(xsrc_in, PFp(base + 0), PFp(base + 1), xsrcn, nsrc, s);
    src_act = xsrcn;
  } else {
    src_act = xdstn;
  }
  lnrun(rn, PFp(base + 2), PFp(base + 3), rn, E, s);  // in-place prenorm of r

  gemm128(xdstn, PFp(base + 4), PFp(base + 5), qbuf, NPL, H, 0, s);   // q
  gemm128(src_act, PFp(base + 6), nullptr, kvbuf, nsrc, H, 0, s);     // k
  gemm128(rn, PFp(base + 9), nullptr, ebuf, E, H, 0, s);              // kr

  fill_u<<<dim3((NPL * NHD + 255) / 256), dim3(256), 0, s>>>(smax, 0x007FFFFFu,
                                                             (long long)NPL * NHD);
  fill_f<<<dim3((NPL * NHD + 255) / 256), dim3(256), 0, s>>>(denb, 0.f,
                                                             (long long)NPL * NHD);
  fill_f<<<dim3((NPL * H + 255) / 256), dim3(256), 0, s>>>(aggb, 0.f,
                                                           (long long)NPL * H);
  sim_kernel<<<dim3((E * NHD + 255) / 256), dim3(256), 0, s>>>(qbuf, kvbuf, ebuf,
                                                               src, dst, smax, simb, E);
  ex_kernel<<<dim3((E * NHD + 255) / 256), dim3(256), 0, s>>>(simb, smax, denb, dst, E);

  gemm128(src_act, PFp(base + 7), PFp(base + 8), kvbuf, nsrc, H, 0, s);  // v
  gemm128(rn, PFp(base + 10), PFp(base + 11), ebuf, E, H, 0, s);         // vr
  agg_kernel<<<dim3((unsigned)(((long long)E * H + 255) / 256)), dim3(256), 0, s>>>(
      kvbuf, ebuf, simb, denb, src, dst, aggb, E);

  cat_kernel<<<dim3((NPL * 2 * H + 255) / 256), dim3(256), 0, s>>>(aggb, xdstn, catb, NPL);
  gemm256(catb, PFp(base + 14), PFp(base + 15), t1, NPL, H, 2, s);   // g = sigmoid
  gemm128(xdstn, PFp(base + 12), PFp(base + 13), t2, NPL, H, 0, s);  // s
  gate_kernel<<<dim3((NPL * H + 255) / 256), dim3(256), 0, s>>>(aggb, t1, t2,
                                                                (long long)NPL * H);
  gemm128(aggb, PFp(base + 16), PFp(base + 17), t1, NPL, H, 0, s);   // o
  lnrun(t1, PFp(base + 18), PFp(base + 19), t1, NPL, s);             // post LN
  add_inplace<<<dim3((NPL * H + 255) / 256), dim3(256), 0, s>>>(xpl, t1,
                                                                (long long)NPL * H);
  lnrun(xpl, PFp(base + 20), PFp(base + 21), t1, NPL, s);            // ffn LN
  gemm128(t1, PFp(base + 22), PFp(base + 23), ffnh, NPL, 4 * H, 1, s);  // relu
  gemm512(ffnh, PFp(base + 24), PFp(base + 25), t1, NPL, H, 0, s);
  add_inplace<<<dim3((NPL * H + 255) / 256), dim3(256), 0, s>>>(xpl, t1,
                                                                (long long)NPL * H);
}

extern "C" void kernel_launch(void* const* d_in, const int* in_sizes, int n_in,
                              void* d_out, int out_size, void* d_ws, size_t ws_size,
                              hipStream_t stream) {
  (void)in_sizes; (void)n_in; (void)out_size; (void)ws_size;
  // input indices (setup_inputs dict insertion order):
  // 0 pos_pt,1 orient_pt,2 pos_pl,3 orient_pl,4 orient_vector_pl,5 x_pt,6 x_pl,
  // 7 map_polygon_batch,8 map_point_batch,9 e_pt2pl,10 e_pl2pl,
  // 11..21 fourier x_pt, 22..32 fourier x_pl, 33..43 fourier r_pt2pl,
  // 44..54 fourier r_pl2pl, 55..82 attn pt2pl(28), 83..108 attn pl2pl(26)
  const int* e1src = PIp(9);
  const int* e1dst = PIp(9) + E1N;
  const int* e2src = PIp(10);
  const int* e2dst = PIp(10) + E2N;

  float* Wb = (float*)d_ws;
  size_t o = 0;
  auto carve = [&](size_t n) { float* p = Wb + o; o += n; return p; };
  float* r1raw = carve((size_t)E1N * 3);
  float* r2raw = carve((size_t)E2N * 3);
  float* remb1 = carve((size_t)E1N * H);
  float* remb2 = carve((size_t)E2N * H);
  float* ebuf  = carve((size_t)E1N * H);
  float* xsrcn = carve((size_t)NPT * H);
  float* kvbuf = carve((size_t)NPT * H);
  float* xpl   = carve((size_t)NPL * H);
  float* xdstn = carve((size_t)NPL * H);
  float* qbuf  = carve((size_t)NPL * H);
  float* aggb  = carve((size_t)NPL * H);
  float* t1    = carve((size_t)NPL * H);
  float* t2    = carve((size_t)NPL * H);
  float* catb  = carve((size_t)NPL * 2 * H);
  float* ffnh  = carve((size_t)NPL * 4 * H);
  float* simb  = carve((size_t)E1N * NHD);
  unsigned* smax = (unsigned*)carve((size_t)NPL * NHD);
  float* denb  = carve((size_t)NPL * NHD);

  float* out_xpt = (float*)d_out;                       // [NPT, H]
  float* out_xpl = out_xpt + (size_t)NPT * H;           // [NPL, T, H]

  // 1) Fourier embeddings (x_pt output goes straight to d_out)
  run_fourier(d_in, 1, PFp(5), 11, out_xpt, NPT, stream);
  run_fourier(d_in, 2, PFp(2), 22, xpl, NPL, stream);   // note: uses pos_pl

  // 2) edge geometry features
  geom_kernel<<<dim3((E1N + 255) / 256), dim3(256), 0, stream>>>(
      PFp(0), PFp(1), PFp(2), PFp(3), PFp(4), e1src, e1dst, r1raw, E1N);
  geom_kernel<<<dim3((E2N + 255) / 256), dim3(256), 0, stream>>>(
      PFp(2), PFp(3), PFp(2), PFp(3), PFp(4), e2src, e2dst, r2raw, E2N);
  run_fourier(d_in, 3, r1raw, 33, remb1, E1N, stream);
  run_fourier(d_in, 3, r2raw, 44, remb2, E2N, stream);

  // 3) attention layers (one of each per reference)
  run_attn_layer(d_in, 55, true, out_xpt, NPT, xpl, remb1, e1src, e1dst, E1N,
                 xsrcn, xdstn, qbuf, kvbuf, ebuf, aggb, t1, t2, catb, ffnh,
                 simb, smax, denb, stream);
  run_attn_layer(d_in, 83, false, xpl, NPL, xpl, remb2, e2src, e2dst, E2N,
                 xsrcn, xdstn, qbuf, kvbuf, ebuf, aggb, t1, t2, catb, ffnh,
                 simb, smax, denb, stream);

  // 4) broadcast x_pl over T
  long long nb = (long long)NPL * TT * H;
  bcast_kernel<<<dim3((unsigned)((nb + 255) / 256)), dim3(256), 0, stream>>>(
      xpl, out_xpl, nb);
}